// EntityMemory_29463475651061
// MI455X (gfx1250) — compile-verified
//
#include <hip/hip_runtime.h>
#include <hip/hip_bf16.h>

// ---------------------------------------------------------------------------
// EntityMemory fused pipeline for gfx1250 (MI455X), wave32.
//  K1: gather span + pseudo = span @ W_f^T + b                  (VALU)
//  K2 (x3): scores = pseudo @ E  via V_WMMA_F32_16X16X4_F32
//           RG rowgroups per block share each B fragment (register) across
//           RG accumulators -> RG x fewer L2 reads of E, RG independent
//           WMMA chains per k-step.
//           mode 0 (RG=2): coarse 8-bit radix histogram (LDS-aggregated)
//           mode 1 (RG=4): fine 8-bit histogram inside critical coarse bin
//           mode 2 (RG=4): collect candidates with 16-bit key >= threshold
//  K3a/K3b: per-row threshold scans
//  K5: bitonic top-k sort of candidates + softmax + picked + proj + scatter
// Workspace use: ~11.6 MB.
// ---------------------------------------------------------------------------

typedef __attribute__((ext_vector_type(2))) float v2f;
typedef __attribute__((ext_vector_type(8))) float v8f;

#define Bb   8
#define Ss   512
#define Dd   768
#define DENT 256
#define Nn   262144
#define Mm   1024
#define CAP  1024
#define KMAX 128

__device__ __forceinline__ unsigned ord32(float f) {
  unsigned u = __float_as_uint(f);
  return (u & 0x80000000u) ? ~u : (u | 0x80000000u);   // monotone float->uint
}

__global__ void zero_u32(unsigned* __restrict__ p, int n) {
  int i = blockIdx.x * blockDim.x + threadIdx.x;
  int stride = gridDim.x * blockDim.x;
  for (; i < n; i += stride) p[i] = 0u;
}

// ---- K1: pseudo embeddings --------------------------------------------------
__global__ __launch_bounds__(DENT)
void pseudo_kernel(const float* __restrict__ X, const float* __restrict__ Wf,
                   const float* __restrict__ bf, const int* __restrict__ pb,
                   const int* __restrict__ pbeg, const int* __restrict__ pend,
                   float* __restrict__ pseudo) {
  __shared__ float span[2 * Dd];
  const int m = blockIdx.x, tid = threadIdx.x;
  const int b = pb[m], beg = pbeg[m], en = pend[m];
  const float* x0 = X + ((size_t)b * Ss + beg) * Dd;
  const float* x1 = X + ((size_t)b * Ss + en) * Dd;
  for (int i = tid; i < 2 * Dd; i += DENT)
    span[i] = (i < Dd) ? x0[i] : x1[i - Dd];
  __syncthreads();
  const float* w = Wf + (size_t)tid * (2 * Dd);
  float acc = bf[tid];
#pragma unroll 8
  for (int j = 0; j < 2 * Dd; ++j) acc = fmaf(span[j], w[j], acc);
  pseudo[m * DENT + tid] = acc;
}

// ---- K2: WMMA scoring GEMM, RG rowgroups/block, 3 modes --------------------
template <int RG, int MODE>
__global__ __launch_bounds__(256)
void score_pass(const float* __restrict__ pseudo, const float* __restrict__ E,
                unsigned* __restrict__ ghistC, unsigned* __restrict__ ghistF,
                const int* __restrict__ Cstar, const unsigned* __restrict__ Tstar,
                unsigned* __restrict__ ccnt, float* __restrict__ cval,
                int* __restrict__ cidx) {
  // A tiles in WMMA pair layout: a_lds[r*4096 + (k>>1)*32 + row*2 + (k&1)]
  __shared__ float    a_lds[RG * DENT * 16];
  __shared__ unsigned lhist[(MODE == 0) ? (RG * 16 * 256) : 1];

  const int tid = threadIdx.x;
  const int wave = tid >> 5, lane = tid & 31;
  const int lane15 = lane & 15, khalf = lane >> 4;
  const int row0 = blockIdx.x * (16 * RG);  // first mention row of this block
  const int cs = blockIdx.y;                // 0..63 column slice (4096 cols)

  for (int i = tid; i < RG * DENT * 16; i += 256) {
    int r = i >> 12;                 // i / 4096
    int j = i & 4095;
    int row = j & 15, k = j >> 4;
    a_lds[r * (DENT * 16) + (k >> 1) * 32 + row * 2 + (k & 1)] =
        pseudo[(size_t)(row0 + r * 16 + row) * DENT + k];
  }
  if (MODE == 0)
    for (int i = tid; i < RG * 16 * 256; i += 256) lhist[i] = 0u;
  __syncthreads();

  // rows this lane-half produces, per rowgroup
  int cst[RG][8]; unsigned tst[RG][8];
#pragma unroll
  for (int r = 0; r < RG; ++r)
#pragma unroll
    for (int vi = 0; vi < 8; ++vi) { cst[r][vi] = 0; tst[r][vi] = 0u; }
  if (MODE == 1) {
#pragma unroll
    for (int r = 0; r < RG; ++r)
#pragma unroll
      for (int vi = 0; vi < 8; ++vi)
        cst[r][vi] = Cstar[row0 + r * 16 + 8 * khalf + vi];
  } else if (MODE == 2) {
#pragma unroll
    for (int r = 0; r < RG; ++r)
#pragma unroll
      for (int vi = 0; vi < 8; ++vi)
        tst[r][vi] = Tstar[row0 + r * 16 + 8 * khalf + vi];
  }

  const int colbase = cs * 4096 + wave * 512;   // 8 waves x 512 cols
  for (int t = 0; t < 32; ++t) {
    const int col = colbase + t * 16 + lane15;  // this lane's output column
    v8f c[RG];
#pragma unroll
    for (int r = 0; r < RG; ++r)
      c[r] = (v8f){0.f, 0.f, 0.f, 0.f, 0.f, 0.f, 0.f, 0.f};
    const float* Ecol = E + (size_t)(2 * khalf) * Nn + col;
#pragma unroll 4
    for (int k = 0; k < DENT; k += 4) {
      // B 4x16: VGPR0 = K k (lo lanes)/k+2 (hi lanes); VGPR1 = k+1/k+3
      const float* e = Ecol + (size_t)k * Nn;
      v2f bv;
      bv.x = e[0];
      bv.y = e[Nn];
      // A 16x4: lanes0-15 = K k,k+1 ; lanes16-31 = K k+2,k+3
      const int apos = (k / 2 + khalf) * 32 + lane15 * 2;
#pragma unroll
      for (int r = 0; r < RG; ++r) {
        v2f a = *(const v2f*)&a_lds[r * (DENT * 16) + apos];
        c[r] = __builtin_amdgcn_wmma_f32_16x16x4_f32(
                   false, a, false, bv, (short)0, c[r], false, false);
      }
    }
    // C/D layout: lane<16 -> N=lane, M=vi ; lane>=16 -> N=lane-16, M=vi+8
#pragma unroll
    for (int r = 0; r < RG; ++r) {
#pragma unroll
      for (int vi = 0; vi < 8; ++vi) {
        float sc = c[r][vi];
        unsigned u = ord32(sc);
        const int rloc = r * 16 + 8 * khalf + vi;   // row within block
        if (MODE == 0) {
          atomicAdd(&lhist[rloc * 256 + (u >> 24)], 1u);
        } else if (MODE == 1) {
          if ((int)(u >> 24) == cst[r][vi])
            atomicAdd(&ghistF[(size_t)(row0 + rloc) * 256 + ((u >> 16) & 255u)],
                      1u);
        } else {
          if ((u >> 16) >= tst[r][vi]) {
            unsigned p = atomicAdd(&ccnt[row0 + rloc], 1u);
            if (p < CAP) {
              cval[(size_t)(row0 + rloc) * CAP + p] = sc;
              cidx[(size_t)(row0 + rloc) * CAP + p] = col;
            }
          }
        }
      }
    }
  }

  if (MODE == 0) {
    __syncthreads();
    for (int i = tid; i < RG * 16 * 256; i += 256) {
      unsigned v = lhist[i];
      if (v)
        atomicAdd(&ghistC[(size_t)(row0 + (i >> 8)) * 256 + (i & 255)], v);
    }
  }
}

// ---- K3a: coarse threshold --------------------------------------------------
__global__ void coarse_select(const unsigned* __restrict__ ghistC,
                              const int* __restrict__ kp,
                              int* __restrict__ Cstar, unsigned* __restrict__ Above) {
  int row = blockIdx.x * blockDim.x + threadIdx.x;
  if (row >= Mm) return;
  int kk = kp[0]; kk = kk < 1 ? 1 : (kk > KMAX ? KMAX : kk);
  unsigned cum = 0u, ab = 0u; int cst = 0;
  for (int b = 255; b >= 0; --b) {
    unsigned h = ghistC[(size_t)row * 256 + b];
    if (cum + h >= (unsigned)kk) { cst = b; ab = cum; break; }
    cum += h;
  }
  Cstar[row] = cst; Above[row] = ab;
}

// ---- K3b: fine threshold ----------------------------------------------------
__global__ void fine_select(const unsigned* __restrict__ ghistF,
                            const unsigned* __restrict__ Above,
                            const int* __restrict__ Cstar,
                            const int* __restrict__ kp,
                            unsigned* __restrict__ Tstar) {
  int row = blockIdx.x * blockDim.x + threadIdx.x;
  if (row >= Mm) return;
  int kk = kp[0]; kk = kk < 1 ? 1 : (kk > KMAX ? KMAX : kk);
  unsigned cum = Above[row]; int fst = 0;
  for (int f = 255; f >= 0; --f) {
    cum += ghistF[(size_t)row * 256 + f];
    if (cum >= (unsigned)kk) { fst = f; break; }
  }
  Tstar[row] = (((unsigned)Cstar[row]) << 8) | (unsigned)fst;
}

// ---- K5: sort candidates, softmax, picked, proj, scatter --------------------
__global__ __launch_bounds__(256)
void finalize_kernel(const float* __restrict__ E, const float* __restrict__ Wb,
                     const float* __restrict__ bb, const int* __restrict__ pb,
                     const int* __restrict__ pbeg, const int* __restrict__ kp,
                     const unsigned* __restrict__ ccnt,
                     const float* __restrict__ cval, const int* __restrict__ cidx,
                     float* __restrict__ y) {
  __shared__ float sval[CAP];
  __shared__ int   sidx[CAP];
  __shared__ float alpha[KMAX];
  __shared__ int   aidx[KMAX];
  __shared__ float red[256];
  __shared__ float picked[DENT];

  const int m = blockIdx.x, tid = threadIdx.x;
  int kk = kp[0]; kk = kk < 1 ? 1 : (kk > KMAX ? KMAX : kk);
  unsigned cnt = ccnt[m]; if (cnt > CAP) cnt = CAP;

  for (int i = tid; i < CAP; i += 256) {
    if (i < (int)cnt) {
      sval[i] = cval[(size_t)m * CAP + i];
      sidx[i] = cidx[(size_t)m * CAP + i];
    } else { sval[i] = -3.4e38f; sidx[i] = 0; }
  }
  // Bitonic sort, descending by value
  for (int size2 = 2; size2 <= CAP; size2 <<= 1)
    for (int stride = size2 >> 1; stride > 0; stride >>= 1) {
      __syncthreads();
      for (int i = tid; i < CAP / 2; i += 256) {
        int pos = 2 * i - (i & (stride - 1));
        int q = pos + stride;
        bool up = (pos & size2) != 0;
        float va = sval[pos], vb = sval[q];
        if (up ? (va > vb) : (va < vb)) {
          sval[pos] = vb; sval[q] = va;
          int ti = sidx[pos]; sidx[pos] = sidx[q]; sidx[q] = ti;
        }
      }
    }
  __syncthreads();
  if ((int)cnt < kk) kk = (int)cnt;
  if (kk < 1) kk = 1;

  const float maxv = sval[0];
  float ev = 0.f;
  if (tid < kk) { ev = expf(sval[tid] - maxv); aidx[tid] = sidx[tid]; }
  red[tid] = ev;
  __syncthreads();
  for (int s = 128; s > 0; s >>= 1) {
    if (tid < s) red[tid] += red[tid + s];
    __syncthreads();
  }
  const float inv = 1.0f / red[0];
  if (tid < kk) alpha[tid] = ev * inv;
  __syncthreads();

  {   // picked[d] = sum_j alpha_j * E[d, idx_j]   (256 threads == DENT)
    float acc = 0.f;
    for (int j = 0; j < kk; ++j)
      acc += alpha[j] * E[(size_t)tid * Nn + aidx[j]];
    picked[tid] = acc;
  }
  __syncthreads();

  const int b = pb[m], beg = pbeg[m];
  float* yo = y + ((size_t)b * Ss + beg) * Dd;
  for (int o = tid; o < Dd; o += 256) {
    const float* w = Wb + (size_t)o * DENT;
    float acc = bb[o];
#pragma unroll 8
    for (int d = 0; d < DENT; ++d) acc = fmaf(picked[d], w[d], acc);
    yo[o] = acc;
  }
}

// ---------------------------------------------------------------------------
extern "C" void kernel_launch(void* const* d_in, const int* in_sizes, int n_in,
                              void* d_out, int out_size, void* d_ws, size_t ws_size,
                              hipStream_t stream) {
  const float* X   = (const float*)d_in[0];
  const float* Wf  = (const float*)d_in[1];
  const float* bf  = (const float*)d_in[2];
  const float* Wb  = (const float*)d_in[3];
  const float* bb  = (const float*)d_in[4];
  const float* E   = (const float*)d_in[5];
  const int*   pb  = (const int*)d_in[6];
  const int*   pbg = (const int*)d_in[7];
  const int*   pnd = (const int*)d_in[8];
  const int*   kp  = (const int*)d_in[9];
  float* y = (float*)d_out;

  // Workspace layout (32-bit units), total ~11.6 MB
  float*    ws     = (float*)d_ws;
  float*    pseudo = ws;                                // 1024*256
  unsigned* ghistC = (unsigned*)(ws + Mm * DENT);       // 1024*256
  unsigned* ghistF = ghistC + Mm * 256;                 // 1024*256
  int*      Cst    = (int*)(ghistF + Mm * 256);         // 1024
  unsigned* Abv    = (unsigned*)(Cst + Mm);             // 1024
  unsigned* Tst    = Abv + Mm;                          // 1024
  unsigned* ccnt   = Tst + Mm;                          // 1024
  float*    cval   = (float*)(ccnt + Mm);               // 1024*CAP
  int*      cidx   = (int*)(cval + (size_t)Mm * CAP);   // 1024*CAP

  zero_u32<<<2048, 256, 0, stream>>>((unsigned*)y, Bb * Ss * Dd);
  zero_u32<<<1024, 256, 0, stream>>>(ghistC, Mm * 256 * 2 + 4 * Mm);

  pseudo_kernel<<<Mm, DENT, 0, stream>>>(X, Wf, bf, pb, pbg, pnd, pseudo);

  // blockIdx.x = rowgroup-block (fast-varying) -> colslice E reuse through L2
  score_pass<2, 0><<<dim3(32, 64), 256, 0, stream>>>(
      pseudo, E, ghistC, ghistF, Cst, Tst, ccnt, cval, cidx);
  coarse_select<<<4, 256, 0, stream>>>(ghistC, kp, Cst, Abv);
  score_pass<4, 1><<<dim3(16, 64), 256, 0, stream>>>(
      pseudo, E, ghistC, ghistF, Cst, Tst, ccnt, cval, cidx);
  fine_select<<<4, 256, 0, stream>>>(ghistF, Abv, Cst, kp, Tst);
  score_pass<4, 2><<<dim3(16, 64), 256, 0, stream>>>(
      pseudo, E, ghistC, ghistF, Cst, Tst, ccnt, cval, cidx);

  finalize_kernel<<<Mm, 256, 0, stream>>>(E, Wb, bb, pb, pbg, kp, ccnt, cval,
                                          cidx, y);
  (void)in_sizes; (void)n_in; (void)out_size; (void)ws_size;
}